// Detector_3418793968106
// MI455X (gfx1250) — compile-verified
//
#include <hip/hip_runtime.h>
#include <cstdint>

// ---------------------------------------------------------------------------
// YOLO-head decode for MI455X (gfx1250).
// Bandwidth-bound (≈245 MB traffic → ~10.5 µs roofline @ 23.3 TB/s).
// Strategy: per-workgroup tile of 32 spatial cells; stage the 270×32 channel
// slab into LDS with CDNA5 async global->LDS loads (ASYNCcnt), then decode
// with coalesced 90-wide row writes. Two-pass decode so exp/sqrt/sigmoid of
// the shared per-row quantities are computed once per anchor row, not per
// output element.
// ---------------------------------------------------------------------------

#define K_TILE 32      // spatial cells per workgroup
#define K_PAD  271     // LDS row stride in floats (coprime with 64 banks)
#define K_NC   270     // channels (3 anchors * 90)
#define K_VEC  90

__device__ __forceinline__ float fast_sigmoid(float x) {
    return 1.0f / (1.0f + __expf(-x));
}

__global__ __launch_bounds__(256) void decode_tile_kernel(
    const float* __restrict__ in,      // [B, 270, HW] for this scale
    float* __restrict__ out,           // full output, rows of 90 floats
    const float* __restrict__ anchors, // [3,2]
    const float* __restrict__ thresh_p,// [1]
    const int* __restrict__ case_p,    // [1]
    int HW, int W, float t, int outRowBase)
{
    __shared__ float sm[K_TILE * K_PAD];   // [cell][channel], padded
    __shared__ float rowv[K_TILE * 3 * 8]; // per-row: n,sp,cx,cy,w,h,scale,mask

    const int b   = blockIdx.y;
    const int s0  = blockIdx.x * K_TILE;
    const int tid = threadIdx.x;

    // ---------------- Stage tile into LDS: async global->LDS ----------------
    // Lane mapping: 32 lanes cover 32 consecutive spatial cells (contiguous
    // 128B per wave per channel); 8 channels per pass across the 8 waves.
    {
        const int lane_s = tid & 31;
        const int cBase  = tid >> 5;                       // 0..7
        const int sIdx   = min(s0 + lane_s, HW - 1);       // clamp keeps EXEC uniform
        #pragma unroll 2
        for (int p = 0; p < 34; ++p) {                     // 34*8 >= 270 channels
            int c  = p * 8 + cBase;
            int cc = min(c, K_NC - 1);
            uint32_t goff = (uint32_t)((((b * K_NC + cc) * HW) + sIdx) * 4);
            uint32_t loff = (uint32_t)(uintptr_t)&sm[lane_s * K_PAD + cc];
            // GVS mode: saddr(64b SGPR base) + vaddr(32b byte offset) -> LDS address in loff
            asm volatile("global_load_async_to_lds_b32 %0, %1, %2"
                         :: "v"(loff), "v"(goff), "s"(in)
                         : "memory");
        }
        asm volatile("s_wait_asynccnt 0x0" ::: "memory");  // wave's async done
    }
    __syncthreads();                                       // visible to all waves

    const float th      = *thresh_p;
    const float invCase = 1.0f / (float)(*case_p);

    // ---------------- Pass A: one thread per anchor-row (96 rows) -----------
    if (tid < K_TILE * 3) {
        int cell = tid / 3;
        int a    = tid - cell * 3;
        int s    = min(s0 + cell, HW - 1);                 // clamped; masked in pass B
        int iy   = s / W;
        int ix   = s - iy * W;
        const float* v = &sm[cell * K_PAD + a * K_VEC];
        float p  = v[0];
        float sp = fast_sigmoid(p);
        float cx = ((float)ix + v[1]) * t;
        float cy = ((float)iy + v[2]) * t;
        float w  = anchors[a * 2 + 0] * __expf(v[3]);
        float h  = anchors[a * 2 + 1] * __expf(v[4]);
        float sc = sqrtf(w * w + h * h) * invCase;
        float* r = &rowv[tid * 8];
        r[0] = (float)b;
        r[1] = sp;
        r[2] = cx;
        r[3] = cy;
        r[4] = w;
        r[5] = h;
        r[6] = sc;
        r[7] = (sp > th) ? 1.0f : 0.0f;
    }
    __syncthreads();

    // ---------------- Pass B: one thread per output element -----------------
    // Consecutive threads -> consecutive output floats: coalesced stores.
    const long long rowBase = (long long)outRowBase + ((long long)b * HW + s0) * 3;
    for (int idx = tid; idx < K_TILE * 3 * K_VEC; idx += 256) {
        int rr   = idx / K_VEC;
        int k    = idx - rr * K_VEC;
        int cell = rr / 3;
        int a    = rr - cell * 3;
        if (s0 + cell >= HW) continue;                     // partial last tile
        const float* r = &rowv[rr * 8];
        float val;
        if (k < 6) {
            val = r[k];                                    // n, sp, cx, cy, w, h
        } else {
            float x = sm[cell * K_PAD + a * K_VEC + k];
            if (k < 18) {
                val = x * r[6];                            // keypoints * scale
            } else {
                int j = k - 18;
                int m = j % 3;                             // triplet: coord,sig,sig
                val = (m == 0) ? (x * r[6]) : fast_sigmoid(x);
            }
        }
        out[(rowBase + rr) * K_VEC + k] = (r[7] != 0.0f) ? val : 0.0f;
    }
}

extern "C" void kernel_launch(void* const* d_in, const int* in_sizes, int n_in,
                              void* d_out, int out_size, void* d_ws, size_t ws_size,
                              hipStream_t stream) {
    const float* out13 = (const float*)d_in[0];
    const float* out26 = (const float*)d_in[1];
    const float* out52 = (const float*)d_in[2];
    const float* an13  = (const float*)d_in[3];
    const float* an26  = (const float*)d_in[4];
    const float* an52  = (const float*)d_in[5];
    const float* thr   = (const float*)d_in[6];
    const int*   casep = (const int*)d_in[7];
    float* out = (float*)d_out;

    const int B = in_sizes[0] / (K_NC * 13 * 13);

    dim3 blk(256);
    auto launch = [&](const float* in, const float* an, int H, float t, int rowBase) {
        int HW = H * H;
        int ntiles = (HW + K_TILE - 1) / K_TILE;
        dim3 grid(ntiles, B);
        decode_tile_kernel<<<grid, blk, 0, stream>>>(in, out, an, thr, casep,
                                                     HW, H, t, rowBase);
    };

    int base13 = 0;
    int base26 = base13 + B * 13 * 13 * 3;
    int base52 = base26 + B * 26 * 26 * 3;

    launch(out13, an13, 13, 32.0f, base13);
    launch(out26, an26, 26, 16.0f, base26);
    launch(out52, an52, 52, 8.0f,  base52);
}